// CAS_new_12446815224568
// MI455X (gfx1250) — compile-verified
//
#include <hip/hip_runtime.h>

typedef __attribute__((ext_vector_type(2))) float v2f;
typedef __attribute__((ext_vector_type(8))) float v8f;

#define VPS   32
#define NCH   128
#define HW    4096
#define K512  512
#define CHW   (NCH*HW)        // 524288 floats per viewport
#define VHW   (VPS*HW)        // 131072 = columns of the raw-reshaped matrix
#define TOTAL (VPS*CHW)       // 16777216 elements (64 MB)

// ---- CDNA5 async global->LDS copies (ASYNCcnt tracked, no VGPR data) ------
__device__ __forceinline__ unsigned lds_off_u32(const void* p) {
    // flat pointer to LDS: low 32 bits are the wave-relative LDS byte offset
    return (unsigned)(unsigned long long)(uintptr_t)p;
}
__device__ __forceinline__ void async_lds_b32(unsigned lds, const float* g) {
    asm volatile("global_load_async_to_lds_b32 %0, %1, off"
                 :: "v"(lds), "v"((unsigned long long)(uintptr_t)g) : "memory");
}
__device__ __forceinline__ void async_lds_b128(unsigned lds, const float* g) {
    asm volatile("global_load_async_to_lds_b128 %0, %1, off"
                 :: "v"(lds), "v"((unsigned long long)(uintptr_t)g) : "memory");
}
__device__ __forceinline__ void async_wait0() {
    asm volatile("s_wait_asynccnt 0x0" ::: "memory");
}

// ---------------------------------------------------------------------------
// Stage A: x_sm[v] = conv_w[128x512] @ cat[v][512x4096] + bias
// cat rows: [0,128)=x1, [128,256)=up2(x2), [256,384)=up4(x3), [384,512)=up4(x4)
// One workgroup = one (v, 16-col) stripe; 8 waves cover the 128 output rows.
// ---------------------------------------------------------------------------
__global__ __launch_bounds__(256) void conv_gemm_kernel(
    const float* __restrict__ x1, const float* __restrict__ x2,
    const float* __restrict__ x3, const float* __restrict__ x4,
    const float* __restrict__ W,  const float* __restrict__ bias,
    float* __restrict__ xsm)
{
    __shared__ float Bs[K512 * 16];   // 32 KB: cat[:, n0:n0+16]
    const int v   = blockIdx.y;
    const int n0  = blockIdx.x * 16;
    const int tid = threadIdx.x;
    const unsigned uBase = lds_off_u32(&Bs[0]);

    // x1 panel: rows 0..127 are contiguous 64B stripes -> async b128
    for (int i = tid; i < 512; i += 256) {            // 2 iters
        int k = i >> 2, ng = (i & 3) * 4;
        async_lds_b128(uBase + (unsigned)(k*16 + ng)*4u,
                       &x1[(v*128 + k)*4096 + n0 + ng]);
    }
    // upsampled panels: nearest-neighbor gather -> async b32
    for (int i = 2048 + tid; i < K512 * 16; i += 256) { // 24 iters
        int k = i >> 4, n = i & 15;
        int hw = n0 + n, h = hw >> 6, w = hw & 63;
        const float* g;
        if (k < 256)      g = &x2[(v*128 + (k - 128))*1024 + (h>>1)*32 + (w>>1)];
        else if (k < 384) g = &x3[(v*128 + (k - 256))*256  + (h>>2)*16 + (w>>2)];
        else              g = &x4[(v*128 + (k - 384))*256  + (h>>2)*16 + (w>>2)];
        async_lds_b32(uBase + (unsigned)i*4u, g);
    }
    async_wait0();
    __syncthreads();

    const int wave = tid >> 5, lane = tid & 31;
    const int row = lane & 15, khalf = lane >> 4;   // A: lanes 0-15 -> K{0,1}, 16-31 -> K{2,3}
    const int o0 = wave * 16;

    v8f acc = {0.f,0.f,0.f,0.f,0.f,0.f,0.f,0.f};
    const float* wrow = W + (o0 + row) * K512 + khalf * 2;
    for (int kt = 0; kt < 128; ++kt) {
        int kb = kt * 4 + khalf * 2;
        v2f a; a.x = wrow[kt*4];          a.y = wrow[kt*4 + 1];
        v2f b; b.x = Bs[kb*16 + row];     b.y = Bs[(kb+1)*16 + row];
        acc = __builtin_amdgcn_wmma_f32_16x16x4_f32(false, a, false, b,
                                                    (short)0, acc, false, false);
    }

    float* outp = xsm + (v*128 + o0) * 4096 + n0 + row;
    #pragma unroll
    for (int i = 0; i < 8; ++i) {
        int m = i + khalf * 8;                       // D vgpr i: rows {i, i+8}
        outp[m * 4096] = acc[i] + bias[o0 + m];
    }
}

// ---------------------------------------------------------------------------
// One 64MB pass: meanV[chw] = mean_v xsm, plus per-(v,ch,chunk) GAP partials.
// grid = 128 ch * 16 chunks of 256 hw.  No atomics -> deterministic.
// ---------------------------------------------------------------------------
__global__ __launch_bounds__(256) void reduce_kernel(
    const float* __restrict__ xsm, float* __restrict__ meanV,
    float* __restrict__ gapPart)
{
    const int tid  = threadIdx.x;
    const int ch   = blockIdx.x >> 4;
    const int chnk = blockIdx.x & 15;
    const int hw   = chnk * 256 + tid;
    const int lane = tid & 31, wid = tid >> 5;
    __shared__ float part[8];

    float s = 0.f;
    for (int v = 0; v < VPS; ++v) {
        float x = xsm[v*CHW + ch*4096 + hw];
        s += x;
        float r = x;
        r += __shfl_down(r, 16, 32);
        r += __shfl_down(r,  8, 32);
        r += __shfl_down(r,  4, 32);
        r += __shfl_down(r,  2, 32);
        r += __shfl_down(r,  1, 32);
        if (lane == 0) part[wid] = r;
        __syncthreads();
        if (tid == 0) {
            float t = part[0]+part[1]+part[2]+part[3]+part[4]+part[5]+part[6]+part[7];
            gapPart[(v*NCH + ch)*16 + chnk] = t;
        }
        __syncthreads();
    }
    meanV[ch*4096 + hw] = s * (1.f / 32.f);
}

// c_q[v,ch] = GAP(x2)[v,ch] = gap/4096 + gamma * mean_v(gap)/4096
__global__ __launch_bounds__(128) void prep_cq_kernel(
    const float* __restrict__ gapPart, const float* __restrict__ gamma,
    float* __restrict__ cq)
{
    __shared__ float gsum[VPS * NCH];   // 16 KB
    const int ch = threadIdx.x;
    for (int v = 0; v < VPS; ++v) {
        float t = 0.f;
        #pragma unroll
        for (int u = 0; u < 16; ++u) t += gapPart[(v*NCH + ch)*16 + u];
        gsum[v*NCH + ch] = t;
    }
    float colsum = 0.f;
    for (int v = 0; v < VPS; ++v) colsum += gsum[v*NCH + ch];
    const float addc = gamma[0] * colsum * (1.f / (32.f * 4096.f));
    for (int v = 0; v < VPS; ++v)
        cq[v*NCH + ch] = gsum[v*NCH + ch] * (1.f / 4096.f) + addc;
}

// C_mat[i,j] = softmax_j(c_q^T c_q)[i,j] * weight[j]; one block per row i.
__global__ __launch_bounds__(128) void cmat_kernel(
    const float* __restrict__ cq, const float* __restrict__ weight,
    float* __restrict__ Cmat)
{
    const int i = blockIdx.x, j = threadIdx.x;
    const int lane = j & 31, wid = j >> 5;
    __shared__ float redm[4], reds[4];

    float s = 0.f;
    for (int v = 0; v < VPS; ++v) s += cq[v*NCH + i] * cq[v*NCH + j];

    float m = s;
    m = fmaxf(m, __shfl_down(m, 16, 32));
    m = fmaxf(m, __shfl_down(m,  8, 32));
    m = fmaxf(m, __shfl_down(m,  4, 32));
    m = fmaxf(m, __shfl_down(m,  2, 32));
    m = fmaxf(m, __shfl_down(m,  1, 32));
    if (lane == 0) redm[wid] = m;
    __syncthreads();
    const float rowmax = fmaxf(fmaxf(redm[0], redm[1]), fmaxf(redm[2], redm[3]));

    const float e = __expf(s - rowmax);
    float t = e;
    t += __shfl_down(t, 16, 32);
    t += __shfl_down(t,  8, 32);
    t += __shfl_down(t,  4, 32);
    t += __shfl_down(t,  2, 32);
    t += __shfl_down(t,  1, 32);
    if (lane == 0) reds[wid] = t;
    __syncthreads();
    const float tot = reds[0] + reds[1] + reds[2] + reds[3];

    Cmat[i*NCH + j] = (e / tot) * weight[j];
}

// ---------------------------------------------------------------------------
// Final: out = C_mat @ M + M with M = raw view [128 x 131072] of
// (xsm + gamma*meanV).  Row c of M needs meanV at ((c&3)*131072 + m); inside
// the k-loop c&3 == khalf*2 (+0/1) and in the epilogue r&3 == i&3, so the
// gamma*meanV term is just 4 per-lane scalars -> LDS tile stages pure xsm
// via async b128, correction folded into the B operands / epilogue.
// ---------------------------------------------------------------------------
__global__ __launch_bounds__(256) void final_gemm_kernel(
    const float* __restrict__ xsm, const float* __restrict__ meanV,
    const float* __restrict__ Cmat, const float* __restrict__ gamma,
    float* __restrict__ out)
{
    __shared__ float Bs[128 * 16];    // 8 KB: xsm[0:128 rows, m0:m0+16]
    const int m0  = blockIdx.x * 16;
    const int tid = threadIdx.x;
    const float g = gamma[0];
    const unsigned uBase = lds_off_u32(&Bs[0]);

    for (int i = tid; i < 512; i += 256) {            // 2 iters, b128 each
        int k = i >> 2, ng = (i & 3) * 4;
        async_lds_b128(uBase + (unsigned)(k*16 + ng)*4u,
                       &xsm[k*VHW + m0 + ng]);
    }

    const int wave = tid >> 5, lane = tid & 31;
    const int row = lane & 15, khalf = lane >> 4;
    const int r0 = wave * 16;

    // per-lane mean-correction scalars: mvE[c] = g * meanV[c*VHW + m0 + row]
    float mvE[4];
    #pragma unroll
    for (int c = 0; c < 4; ++c) mvE[c] = g * meanV[c*VHW + m0 + row];
    const float mv0 = mvE[khalf*2], mv1 = mvE[khalf*2 + 1];

    async_wait0();
    __syncthreads();

    v8f acc = {0.f,0.f,0.f,0.f,0.f,0.f,0.f,0.f};
    const float* arow = Cmat + (r0 + row) * NCH + khalf * 2;
    for (int kt = 0; kt < 32; ++kt) {
        int kb = kt * 4 + khalf * 2;
        v2f a; a.x = arow[kt*4];            a.y = arow[kt*4 + 1];
        v2f b; b.x = Bs[kb*16 + row] + mv0; b.y = Bs[(kb+1)*16 + row] + mv1;
        acc = __builtin_amdgcn_wmma_f32_16x16x4_f32(false, a, false, b,
                                                    (short)0, acc, false, false);
    }

    #pragma unroll
    for (int i = 0; i < 8; ++i) {
        int r = r0 + i + khalf * 8;
        out[r*VHW + m0 + row] = acc[i] + Bs[r*16 + row] + mvE[i & 3];
    }
}

// ---------------------------------------------------------------------------
extern "C" void kernel_launch(void* const* d_in, const int* in_sizes, int n_in,
                              void* d_out, int out_size, void* d_ws, size_t ws_size,
                              hipStream_t stream)
{
    (void)in_sizes; (void)n_in; (void)out_size; (void)ws_size;
    const float* x1     = (const float*)d_in[0];
    const float* x2     = (const float*)d_in[1];
    const float* x3     = (const float*)d_in[2];
    const float* x4     = (const float*)d_in[3];
    const float* conv_w = (const float*)d_in[4];
    const float* conv_b = (const float*)d_in[5];
    const float* gamma  = (const float*)d_in[6];
    const float* weight = (const float*)d_in[7];
    // d_in[8] (lin_w) is mathematically irrelevant: softmax over constant rows
    // is exactly uniform, so the viewport attention reduces to a batch mean.

    float* ws      = (float*)d_ws;
    float* xsm     = ws;                       // 16,777,216 f
    float* meanV   = xsm   + TOTAL;            //    524,288 f
    float* gapPart = meanV + CHW;              //     65,536 f
    float* cq      = gapPart + VPS*NCH*16;     //      4,096 f
    float* Cmat    = cq      + VPS*NCH;        //     16,384 f
    float* outp    = (float*)d_out;

    hipLaunchKernelGGL(conv_gemm_kernel, dim3(HW/16, VPS), dim3(256), 0, stream,
                       x1, x2, x3, x4, conv_w, conv_b, xsm);
    hipLaunchKernelGGL(reduce_kernel, dim3(NCH*16), dim3(256), 0, stream,
                       xsm, meanV, gapPart);
    hipLaunchKernelGGL(prep_cq_kernel, dim3(1), dim3(128), 0, stream,
                       gapPart, gamma, cq);
    hipLaunchKernelGGL(cmat_kernel, dim3(NCH), dim3(128), 0, stream,
                       cq, weight, Cmat);
    hipLaunchKernelGGL(final_gemm_kernel, dim3(VHW/16), dim3(256), 0, stream,
                       xsm, meanV, Cmat, gamma, outp);
}